// Jitter_84765474553865
// MI455X (gfx1250) — compile-verified
//
#include <hip/hip_runtime.h>
#include <stdint.h>

// Jitter gather: out[row, t] = x[row, g[t]], g[t] in {t-1, t, t+1} with boundary rules
// (t==0 -> 1, t==T-1 -> T-2), replacement decided per-t by replace_mask.
//
// 0 FLOPs, ~524 MB traffic -> ~22.5us floor at 23.3 TB/s. Strategy: stage one whole
// row (T=4000 floats, 16KB) into LDS per block via GLOBAL_LOAD_ASYNC_TO_LDS_B128
// (ASYNCcnt path, no VGPR round-trip), then resolve the +/-1 gather entirely from
// LDS and write back coalesced 128-bit stores. No halos, every x element read once.

#define THREADS   256
#define MAX_T     4096   // LDS row buffer capacity (T=4000 fits)

__global__ __launch_bounds__(THREADS)
void Jitter_gather_kernel(const float* __restrict__ x,
                          const unsigned char* __restrict__ replace_mask,
                          const int* __restrict__ neighbor_bits,
                          float* __restrict__ out,
                          int T)
{
    __shared__ __align__(16) float tile[MAX_T];   // whole row, 16KB for T=4000

    const long long rowBase = (long long)blockIdx.x * (long long)T;
    const float*    src     = x   + rowBase;
    float*          dst     = out + rowBase;
    const int       nChunks = T >> 2;             // T % 4 == 0 (T = 4000)

    // ---- Stage the row: each lane async-copies 16B chunks global -> LDS.
    for (int c = threadIdx.x; c < nChunks; c += THREADS) {
        unsigned ldsAddr = (unsigned)(uintptr_t)(&tile[c * 4]);
        asm volatile("global_load_async_to_lds_b128 %0, %1, off"
                     :
                     : "v"(ldsAddr),
                       "v"((unsigned long long)(uintptr_t)(src + c * 4))
                     : "memory");
    }

    // Wait for this wave's async transfers to land in LDS, then make them
    // visible to all waves in the workgroup.
    asm volatile("s_wait_asynccnt 0" ::: "memory");
    __syncthreads();

    // ---- Gather 4 consecutive t per chunk out of LDS, store one float4.
    for (int c = threadIdx.x; c < nChunks; c += THREADS) {
        const int t0 = c * 4;
        const int4   nb4 = *reinterpret_cast<const int4*>(neighbor_bits + t0);
        const uchar4 m4  = *reinterpret_cast<const uchar4*>(replace_mask + t0);

        const int           nbv[4] = { nb4.x, nb4.y, nb4.z, nb4.w };
        const unsigned char mv[4]  = { m4.x, m4.y, m4.z, m4.w };

        float r[4];
#pragma unroll
        for (int j = 0; j < 4; ++j) {
            const int t   = t0 + j;
            const int off = (nbv[j] > 0) ? 1 : -1;
            const int nb  = (t == 0) ? 1 : ((t == T - 1) ? (T - 2) : (t + off));
            const int g   = mv[j] ? nb : t;        // g in [0, T-1] always
            r[j] = tile[g];
        }

        float4 v; v.x = r[0]; v.y = r[1]; v.z = r[2]; v.w = r[3];
        *reinterpret_cast<float4*>(dst + t0) = v;
    }
}

extern "C" void kernel_launch(void* const* d_in, const int* in_sizes, int n_in,
                              void* d_out, int out_size, void* d_ws, size_t ws_size,
                              hipStream_t stream) {
    const float*         x    = (const float*)d_in[0];          // [B, C, T] f32
    const unsigned char* mask = (const unsigned char*)d_in[1];  // [T] bool (1 byte each)
    const int*           bits = (const int*)d_in[2];            // [T] int32
    float*               outp = (float*)d_out;

    const int T = in_sizes[2];                       // 4000
    const long long total = (long long)in_sizes[0];  // B*C*T
    const int rows = (int)(total / T);               // B*C = 16384

    Jitter_gather_kernel<<<dim3((unsigned)rows), THREADS, 0, stream>>>(
        x, mask, bits, outp, T);
}